// cft_attention_30004641530188
// MI455X (gfx1250) — compile-verified
//
#include <hip/hip_runtime.h>
#include <hip/hip_bf16.h>
#include <math.h>

// Problem shape (validated at launch from in_sizes)
#define DIM 128

typedef __attribute__((ext_vector_type(2))) float v2f;
typedef __attribute__((ext_vector_type(8))) float v8f;

// ---------------------------------------------------------------------------
// K0: zero the accumulation buffers (out and segment-sum s) every call.
// ---------------------------------------------------------------------------
__global__ void zero_kernel(float* __restrict__ out, float* __restrict__ s,
                            int nout, int ns) {
    int i = blockIdx.x * blockDim.x + threadIdx.x;
    int stride = gridDim.x * blockDim.x;
    for (int j = i; j < nout; j += stride) out[j] = 0.0f;
    for (int j = i; j < ns;  j += stride) s[j]  = 0.0f;
}

// ---------------------------------------------------------------------------
// K1: layernorm, one wave32 per row. 128 floats = 1 float4 per lane.
// ---------------------------------------------------------------------------
__global__ void layernorm_kernel(const float* __restrict__ embs,
                                 float* __restrict__ xn, int n) {
    int wave = (int)((blockIdx.x * blockDim.x + threadIdx.x) >> 5);
    int lane = threadIdx.x & 31;
    if (wave >= n) return;

    const float4 v = *(const float4*)(embs + (size_t)wave * DIM + lane * 4);
    float sum = v.x + v.y + v.z + v.w;
    float sq  = v.x * v.x + v.y * v.y + v.z * v.z + v.w * v.w;
    #pragma unroll
    for (int off = 16; off > 0; off >>= 1) {
        sum += __shfl_xor(sum, off, 32);
        sq  += __shfl_xor(sq,  off, 32);
    }
    const float m   = sum * (1.0f / DIM);
    const float var = sq * (1.0f / DIM) - m * m;
    const float r   = rsqrtf(var + 1e-5f);

    float4 o;
    o.x = (v.x - m) * r; o.y = (v.y - m) * r;
    o.z = (v.z - m) * r; o.w = (v.w - m) * r;
    *(float4*)(xn + (size_t)wave * DIM + lane * 4) = o;
}

// ---------------------------------------------------------------------------
// K2: 16 edges per wave. Dots computed as the diagonal of a 16x16 WMMA
// accumulation: 32 chained V_WMMA_F32_16X16X4_F32 over K=128.
//
// A (16x4 f32) layout: lanes 0-15 hold (M=lane, K=k+0/k+1) in regs 0/1,
// lanes 16-31 hold (M=lane-16, K=k+2/k+3). B (4x16) mirrors this with
// N=lane&15. So every lane feeds matrix index (lane&15) and K-offset
// (lane<16 ? 0,1 : 2,3) -> one aligned float2 gather per operand per step.
// Diagonal C[m][m]: m=0..7 -> reg m, lane m; m=8..15 -> reg m-8, lane m+16.
// ---------------------------------------------------------------------------
__global__ void edge_attn_wmma_kernel(const float* __restrict__ xn,
                                      const int* __restrict__ spe,
                                      float* __restrict__ ew,
                                      float* __restrict__ s,
                                      long nE) {
    long wave = (long)(blockIdx.x * blockDim.x + threadIdx.x) >> 5;
    int  lane = threadIdx.x & 31;
    long base = wave * 16;
    if (base >= nE) return;                       // uniform per wave

    const int  m  = lane & 15;                    // which edge of the batch
    const long e  = base + m;
    const long ec = (e < nE) ? e : (nE - 1);      // clamp tail (masked later)
    const int  r  = spe[ec * 3 + 0];
    const int  c  = spe[ec * 3 + 1];

    const int koff = (lane >= 16) ? 2 : 0;
    const float* __restrict__ arow = xn + (size_t)r * DIM + koff;
    const float* __restrict__ brow = xn + (size_t)c * DIM + koff;

    v8f acc = {0.f, 0.f, 0.f, 0.f, 0.f, 0.f, 0.f, 0.f};
    #pragma unroll
    for (int k = 0; k < DIM; k += 4) {
        const float2 af = *(const float2*)(arow + k);
        const float2 bf = *(const float2*)(brow + k);
        v2f a; a.x = af.x; a.y = af.y;
        v2f b; b.x = bf.x; b.y = bf.y;
        acc = __builtin_amdgcn_wmma_f32_16x16x4_f32(
            /*neg_a=*/false, a, /*neg_b=*/false, b,
            /*c_mod=*/(short)0, acc, /*reuse_a=*/false, /*reuse_b=*/false);
    }

    // Pull the diagonal element this lane owns (lanes 0-7 and 24-31).
    const bool active = (lane < 8) || (lane >= 24);
    float dot = 0.0f;
    #pragma unroll
    for (int v = 0; v < 8; ++v)
        if ((lane & 7) == v) dot = acc[v];

    if (active && e < nE) {
        const float attn = dot * 0.08838834764831845f;   // 1/sqrt(128)
        const float w = fminf(__expf(attn), 5.0f);       // clip(exp, -5, 5)
        ew[e] = w;
        unsafeAtomicAdd(&s[r], w);                       // global_atomic_add_f32
    }
}

// ---------------------------------------------------------------------------
// K3: one wave per edge: out[row] += (e/s[row]) * x[col]. float4 gather,
// 4 hardware f32 atomics per lane; everything L2-resident.
// ---------------------------------------------------------------------------
__global__ void edge_scatter_kernel(const float* __restrict__ xn,
                                    const int* __restrict__ spe,
                                    const float* __restrict__ ew,
                                    const float* __restrict__ s,
                                    float* __restrict__ out, long nE) {
    long e = (long)(blockIdx.x * blockDim.x + threadIdx.x) >> 5;
    int lane = threadIdx.x & 31;
    if (e >= nE) return;

    const int r = spe[e * 3 + 0];
    const int c = spe[e * 3 + 1];
    const float w = ew[e] / s[r];   // s[r] >= ew[e] > 0 whenever this edge exists

    const float4 v = *(const float4*)(xn + (size_t)c * DIM + lane * 4);
    float* dst = out + (size_t)r * DIM + lane * 4;
    unsafeAtomicAdd(dst + 0, v.x * w);
    unsafeAtomicAdd(dst + 1, v.y * w);
    unsafeAtomicAdd(dst + 2, v.z * w);
    unsafeAtomicAdd(dst + 3, v.w * w);
}

// ---------------------------------------------------------------------------
extern "C" void kernel_launch(void* const* d_in, const int* in_sizes, int n_in,
                              void* d_out, int out_size, void* d_ws, size_t ws_size,
                              hipStream_t stream) {
    const float* embs = (const float*)d_in[0];
    const int*   spe  = (const int*)d_in[1];
    float*       out  = (float*)d_out;

    const int  n  = in_sizes[0] / DIM;   // 50000
    const long nE = in_sizes[1] / 3;     // 625000

    // Workspace: xn (n*128 f32) | s (n f32) | ew (E f32)  ~= 28.3 MB
    float* xn = (float*)d_ws;
    float* s  = xn + (size_t)n * DIM;
    float* ew = s + n;

    // K0: zero accumulators (must happen every call; harness replays graph).
    zero_kernel<<<512, 256, 0, stream>>>(out, s, n * DIM, n);

    // K1: layernorm, 8 waves (rows) per 256-thread block.
    layernorm_kernel<<<(n + 7) / 8, 256, 0, stream>>>(embs, xn, n);

    // K2: WMMA edge dots + exp + segment-sum. 16 edges/wave, 8 waves/block.
    const long waves2 = (nE + 15) / 16;
    edge_attn_wmma_kernel<<<(int)((waves2 + 7) / 8), 256, 0, stream>>>(
        xn, spe, ew, s, nE);

    // K3: scatter accumulate. 1 edge/wave, 8 waves/block.
    edge_scatter_kernel<<<(int)((nE + 7) / 8), 256, 0, stream>>>(
        xn, spe, ew, s, out, nE);
}